// IoU_31336081391713
// MI455X (gfx1250) — compile-verified
//
#include <hip/hip_runtime.h>

#define YOLO_I 20.0f
#define BPB 256            // boxes per block
#define ANCH 5
#define RES_CHUNKS ((BPB * 5 * 4) / 16)    // 320 x 16B for result tile

typedef unsigned v4u __attribute__((ext_vector_type(4)));
typedef unsigned v8u __attribute__((ext_vector_type(8)));

// ---- helpers ----------------------------------------------------------------
__device__ __forceinline__ unsigned lds_off(const void* p) {
    // generic LDS pointer: low 32 bits are the wave-relative LDS byte address
    return (unsigned)(unsigned long long)p;
}
__device__ __forceinline__ unsigned rfl(unsigned x) {
    return (unsigned)__builtin_amdgcn_readfirstlane((int)x);
}

// ---- CDNA5 Tensor Data Mover: 1-D contiguous tile, 4-byte elements ----------
// D# group0: [1:0]=count=1 | [63:32]=lds_addr | [120:64]=global_addr | [127:126]=type=2
// D# group1: [17:16]=data_size=2 (4B) | [79:48]=tensor_dim0 | [111:80]=tensor_dim1=1
//            [127:112]=tile_dim0 | [207:160]=tensor_dim0_stride | rest 0
__device__ __forceinline__ void tdm_copy_1d(unsigned lds_addr, const void* gaddr,
                                            unsigned n_dwords, bool is_store) {
    const unsigned long long ga = (unsigned long long)gaddr;
    const unsigned nd = rfl(n_dwords);
    v4u g0;
    g0[0] = 1u;                                            // count=1, user descriptor
    g0[1] = rfl(lds_addr);                                 // lds_addr
    g0[2] = rfl((unsigned)ga);                             // global_addr[31:0]
    g0[3] = (rfl((unsigned)(ga >> 32)) & 0x01FFFFFFu)      // global_addr[56:32]
            | 0x80000000u;                                 // type=2 ("image")
    v8u g1;
    g1[0] = 0x00020000u;                                   // data_size=4B, wg_mask=0
    g1[1] = (nd & 0xFFFFu) << 16;                          // tensor_dim0[15:0]
    g1[2] = ((nd >> 16) & 0xFFFFu) | (1u << 16);           // tensor_dim0[31:16], tensor_dim1=1
    g1[3] = (nd & 0xFFFFu) << 16;                          // tile_dim0 = n_dwords
    g1[4] = 0u;                                            // tile_dim1/2 unused
    g1[5] = nd;                                            // tensor_dim0_stride[31:0]
    g1[6] = 0u;                                            // stride hi / dim1_stride
    g1[7] = 0u;
    if (is_store)
        asm volatile("tensor_store_from_lds %0, %1" :: "s"(g0), "s"(g1) : "memory");
    else
        asm volatile("tensor_load_to_lds %0, %1" :: "s"(g0), "s"(g1) : "memory");
}

// ---- CDNA5 async LDS->global store (ASYNCcnt-tracked) -----------------------
__device__ __forceinline__ void async_store16(void* gaddr, unsigned lds_addr) {
    asm volatile("global_store_async_from_lds_b128 %0, %1, off"
                 :: "v"((unsigned long long)gaddr), "v"(lds_addr) : "memory");
}
__device__ __forceinline__ void wait_async0() {
    asm volatile("s_wait_asynccnt 0x0" ::: "memory");
}

// ---- per-box IoU math -------------------------------------------------------
__device__ __forceinline__ void iou_box(float ci, float cj,
                                        const float* __restrict__ orow, // 25 floats
                                        const float* __restrict__ trow, // 5 floats
                                        float* __restrict__ rrow)       // 5 floats
{
    const float yi = YOLO_I;
    const float cic = ci * yi + 10.0f;
    const float cjc = cj * yi + 10.0f;

    const float dxg = trow[1], dyg = trow[2];
    const float hg = trow[3] * yi, wg = trow[4] * yi;
    const float xcg = cic + dxg * yi;
    const float ycg = cjc + dyg * yi;
    const float gx1 = ycg - wg * 0.5f, gy1 = xcg - hg * 0.5f;
    const float gx2 = ycg + wg * 0.5f, gy2 = xcg + hg * 0.5f;
    const float area_g = (gx2 - gx1) * (gy2 - gy1);

#pragma unroll
    for (int a = 0; a < ANCH; ++a) {
        const float dx = orow[a * 5 + 1], dy = orow[a * 5 + 2];
        const float h = orow[a * 5 + 3] * yi, w = orow[a * 5 + 4] * yi;
        const float xc = cic + dx * yi;
        const float yc = cjc + dy * yi;
        const float px1 = yc - w * 0.5f, py1 = xc - h * 0.5f;
        const float px2 = yc + w * 0.5f, py2 = xc + h * 0.5f;
        const float area_p = (px2 - px1) * (py2 - py1);
        const float ltx = fmaxf(px1, gx1), lty = fmaxf(py1, gy1);
        const float rbx = fminf(px2, gx2), rby = fminf(py2, gy2);
        const float ww = fmaxf(rbx - ltx, 0.0f);
        const float hh = fmaxf(rby - lty, 0.0f);
        const float inter = ww * hh;
        const float uni = area_p + area_g - inter;
        rrow[a] = inter / uni;
    }
}

__global__ void __launch_bounds__(BPB)
iou_kernel(const int* __restrict__ cells, const float* __restrict__ outp,
           const float* __restrict__ tgt, float* __restrict__ res, int n)
{
    __shared__ float s_outp[BPB * 25];   // 25.6 KB
    __shared__ float s_tgt [BPB * 5];    //  5.0 KB
    __shared__ int   s_cell[BPB * 2];    //  2.0 KB
    __shared__ float s_res [BPB * 5];    //  5.0 KB

    const int tid = threadIdx.x;
    const long long base = (long long)blockIdx.x * BPB;
    const bool full = (base + BPB) <= (long long)n;   // block-uniform

    if (full && tid < 32) {
        // ---- wave 0: three descriptor-driven TDM loads stage all input tiles ----
        tdm_copy_1d(lds_off(s_outp), outp + base * 25, BPB * 25, false); // 25.6 KB
        tdm_copy_1d(lds_off(s_tgt),  tgt  + base * 5,  BPB * 5,  false); //  5.0 KB
        tdm_copy_1d(lds_off(s_cell), cells + base * 2, BPB * 2,  false); //  2.0 KB
        __builtin_amdgcn_s_wait_tensorcnt(0);
    }
    __syncthreads();                   // TDM-written tiles visible to all waves

    if (full) {
        // LDS reads: strides 25/5/2 dwords -> bank-conflict-free on 64 banks
        const float ci = (float)s_cell[tid * 2 + 0];
        const float cj = (float)s_cell[tid * 2 + 1];
        iou_box(ci, cj, &s_outp[tid * 25], &s_tgt[tid * 5], &s_res[tid * 5]);

        __syncthreads();               // s_res complete across the block

        // ---- stream results out with async 128-bit LDS->global stores ----
        char* g_res = (char*)(res + base * 5);
        const unsigned l_res = lds_off(s_res);
        for (int i = tid; i < RES_CHUNKS; i += BPB)
            async_store16(g_res + (size_t)i * 16, l_res + (unsigned)i * 16u);
        wait_async0();
    } else {
        // tail block (not taken for N=4M, kept for generality): scalar path
        const long long box = base + tid;
        if (box < n) {
            float orow[25], trow[5], rrow[5];
#pragma unroll
            for (int k = 0; k < 25; ++k) orow[k] = outp[box * 25 + k];
#pragma unroll
            for (int k = 0; k < 5; ++k) trow[k] = tgt[box * 5 + k];
            iou_box((float)cells[box * 2 + 0], (float)cells[box * 2 + 1],
                    orow, trow, rrow);
#pragma unroll
            for (int k = 0; k < 5; ++k) res[box * 5 + k] = rrow[k];
        }
    }
}

extern "C" void kernel_launch(void* const* d_in, const int* in_sizes, int n_in,
                              void* d_out, int out_size, void* d_ws, size_t ws_size,
                              hipStream_t stream) {
    (void)n_in; (void)out_size; (void)d_ws; (void)ws_size;
    const int*   cells = (const int*)d_in[0];    // [N,2] int32
    const float* outp  = (const float*)d_in[1];  // [N,5,5] f32
    const float* tgt   = (const float*)d_in[2];  // [N,5] f32
    float*       res   = (float*)d_out;          // [N,5] f32

    const int n = in_sizes[0] / 2;               // N boxes
    const int blocks = (n + BPB - 1) / BPB;
    iou_kernel<<<blocks, BPB, 0, stream>>>(cells, outp, tgt, res, n);
}